// SinkhornKnoppCenteringModule_35167192219712
// MI455X (gfx1250) — compile-verified
//
#include <hip/hip_runtime.h>

#define K_DIM 4096
#define B_DIM 16384
#define NB_CHUNKS 64
#define ROWS_PER_CHUNK (B_DIM / NB_CHUNKS)  // 256 rows per pass-A block

// exp(s/0.05) = exp2(s * 20 * log2(e))
__device__ __forceinline__ float expeps(float s) {
  return __builtin_amdgcn_exp2f(s * 28.853900817779268f);
}

typedef float v2f __attribute__((ext_vector_type(2)));
typedef float v8f __attribute__((ext_vector_type(8)));

// ---------------------------------------------------------------------------
// Pass A: t_k = sum_b E[b,k] * w_b  (column reduction over the batch dim)
// via V_WMMA_F32_16X16X4_F32:  D(16x16) = A(16x4) x B(4x16) + C
//   B operand = E tile: B[kappa, n] = exp(scores[b0+kappa][k0+n])
//   A operand = w broadcast over M: A[m, kappa] = w[b0+kappa]
// => every row of D accumulates t-partials for k0..k0+15.
// Layouts (ISA 7.12.2): A vgpr j, lane l -> (M=l&15, K=j+2*(l>>4));
// B symmetric: vgpr j, lane l -> (K=j+2*(l>>4), N=l&15).
// Each block: 8 waves x 16 k-cols = 128 k, over a 256-row batch chunk.
// Writes exclusive partials -> deterministic, no atomics.
// ---------------------------------------------------------------------------
__global__ __launch_bounds__(256) void pass_a_wmma(
    const float* __restrict__ scores, const float* __restrict__ w,
    float* __restrict__ tpart) {
  const int lane = threadIdx.x & 31;
  const int wave = threadIdx.x >> 5;
  const int k0 = blockIdx.x * 128 + wave * 16;
  const int n = lane & 15;
  const int koff = (lane >> 4) << 1;  // 0 for lanes 0-15, 2 for lanes 16-31
  const int bbase = blockIdx.y * ROWS_PER_CHUNK;

  const float* srow = scores + (size_t)(bbase + koff) * K_DIM + k0 + n;
  const float* wrow = w + bbase + koff;

  v8f acc = {};
#pragma unroll 4
  for (int b = 0; b < ROWS_PER_CHUNK; b += 4) {
    const float* p = srow + (size_t)b * K_DIM;
    v2f Bv, Av;
    Bv.x = expeps(p[0]);      // (K=koff,   N=n)
    Bv.y = expeps(p[K_DIM]);  // (K=koff+1, N=n)
    Av.x = wrow[b];           // (M=*, K=koff)
    Av.y = wrow[b + 1];       // (M=*, K=koff+1)
    acc = __builtin_amdgcn_wmma_f32_16x16x4_f32(
        /*neg_a=*/false, Av, /*neg_b=*/false, Bv,
        /*c_mod=*/(short)0, acc, /*reuse_a=*/false, /*reuse_b=*/false);
  }
  // acc[0], lanes 0-15 hold (M=0, N=lane) = t-partial for k0+lane
  if (lane < 16)
    tpart[(size_t)blockIdx.y * K_DIM + k0 + lane] = acc[0];
}

// u_k = r / sum_chunks tpart[chunk][k]
__global__ __launch_bounds__(256) void reduce_u_kernel(
    const float* __restrict__ tpart, float* __restrict__ u) {
  const int k = blockIdx.x * 256 + threadIdx.x;
  float t = 0.f;
#pragma unroll
  for (int cb = 0; cb < NB_CHUNKS; ++cb) t += tpart[(size_t)cb * K_DIM + k];
  u[k] = (1.0f / K_DIM) / t;
}

__global__ void init_ones(float* __restrict__ v) {
  v[blockIdx.x * 256 + threadIdx.x] = 1.0f;
}

// ---------------------------------------------------------------------------
// Pass B: s_b = sum_k E[b,k]*u_k (row reduction). One block per row (16 KB,
// contiguous, float4 loads). it<2: v_b = c/s_b.  Final: re-read the (cache
// hot) row and emit out[b,k] = E[b,k]*u_k / s_b.
// ---------------------------------------------------------------------------
template <bool FINAL>
__global__ __launch_bounds__(256) void pass_b(
    const float* __restrict__ scores, const float* __restrict__ u,
    float* __restrict__ v, float* __restrict__ out) {
  __shared__ float red[256];
  const int tid = threadIdx.x;
  const size_t base = (size_t)blockIdx.x * K_DIM;

  float acc = 0.f;
#pragma unroll
  for (int i = 0; i < K_DIM / 1024; ++i) {
    const int idx = i * 1024 + tid * 4;
    const float4 s = *(const float4*)(scores + base + idx);
    const float4 uu = *(const float4*)(u + idx);
    acc += expeps(s.x) * uu.x + expeps(s.y) * uu.y + expeps(s.z) * uu.z +
           expeps(s.w) * uu.w;
  }
  red[tid] = acc;
  __syncthreads();
  for (int off = 128; off > 0; off >>= 1) {
    if (tid < off) red[tid] += red[tid + off];
    __syncthreads();
  }
  const float stot = red[0];

  if (FINAL) {
    const float inv = 1.0f / stot;
#pragma unroll
    for (int i = 0; i < K_DIM / 1024; ++i) {
      const int idx = i * 1024 + tid * 4;
      const float4 s = *(const float4*)(scores + base + idx);
      const float4 uu = *(const float4*)(u + idx);
      float4 o;
      o.x = expeps(s.x) * uu.x * inv;
      o.y = expeps(s.y) * uu.y * inv;
      o.z = expeps(s.z) * uu.z * inv;
      o.w = expeps(s.w) * uu.w * inv;
      *(float4*)(out + base + idx) = o;
    }
  } else {
    if (tid == 0) v[blockIdx.x] = (1.0f / B_DIM) / stot;
  }
}

extern "C" void kernel_launch(void* const* d_in, const int* in_sizes, int n_in,
                              void* d_out, int out_size, void* d_ws,
                              size_t ws_size, hipStream_t stream) {
  (void)in_sizes; (void)n_in; (void)out_size; (void)ws_size;
  const float* scores = (const float*)d_in[0];
  float* out = (float*)d_out;

  // Workspace layout (floats): v[16384] | u[4096] | tpart[64*4096]  (~1.1 MB)
  float* v = (float*)d_ws;
  float* u = v + B_DIM;
  float* tpart = u + K_DIM;

  init_ones<<<B_DIM / 256, 256, 0, stream>>>(v);  // v0 = 1

  for (int it = 0; it < 3; ++it) {
    pass_a_wmma<<<dim3(K_DIM / 128, NB_CHUNKS), 256, 0, stream>>>(scores, v,
                                                                  tpart);
    reduce_u_kernel<<<K_DIM / 256, 256, 0, stream>>>(tpart, u);
    if (it < 2)
      pass_b<false><<<B_DIM, 256, 0, stream>>>(scores, u, v, out);
    else
      pass_b<true><<<B_DIM, 256, 0, stream>>>(scores, u, v, out);
  }
}